// FCR_3891240370733
// MI455X (gfx1250) — compile-verified
//
#include <hip/hip_runtime.h>
#include <hip/hip_bf16.h>
#include <math.h>

// FCR contrastive loss: 2D DFT via V_WMMA_F32_16X16X4_F32 matmuls + fused reductions.
// B=32, C=3, H=W=256, K(neg)=2. Output: single float.

typedef __attribute__((ext_vector_type(2))) float v2f;
typedef __attribute__((ext_vector_type(8))) float v8f;

#define N256 256
#define PLANE_ELEMS (N256 * N256)          // 65536
#define PLANES_PER_TENSOR 96               // B*C = 32*3
#define TOTAL_PLANES 288                   // 3 tensors
#define TILES_PER_PLANE 256                // (256/16)^2
#define PART_PER_BATCH 768                 // C * TILES_PER_PLANE
#define EPS_F 1e-7f

static __device__ __forceinline__ v8f wmma_f32(v2f a, v2f b, v8f c) {
    // f32 variant: NEG/NEG_HI must be zero (only C-modifiers exist per ISA).
    return __builtin_amdgcn_wmma_f32_16x16x4_f32(false, a, false, b,
                                                 (short)0, c, false, false);
}

// ---------------------------------------------------------------------------
// Kernel 0: DFT twiddle matrix F[k][n] = exp(-2*pi*i*k*n/256), split re/im.
// Exact argument reduction: angle depends only on (k*n) mod 256.
// ---------------------------------------------------------------------------
__global__ void fcr_twiddles(float* __restrict__ Fr, float* __restrict__ Fi) {
    int k = blockIdx.x;
    int n = threadIdx.x;
    int t = (k * n) & 255;
    float theta = -6.283185307179586f * (float)t * (1.0f / 256.0f);
    float s, c;
    sincosf(theta, &s, &c);
    Fr[k * N256 + n] = c;
    Fi[k * N256 + n] = s;
}

// ---------------------------------------------------------------------------
// Kernel 1: row transform Z = F * X  (X real 256x256 plane -> Zr,Zi).
// grid = (32, 288), block = 256 (8 waves); one wave per 16x16 output tile.
// ---------------------------------------------------------------------------
__global__ void fcr_stage1(const float* __restrict__ Xa,
                           const float* __restrict__ Xp,
                           const float* __restrict__ Xn,
                           const float* __restrict__ Fr,
                           const float* __restrict__ Fi,
                           float* __restrict__ Zr,
                           float* __restrict__ Zi) {
    const int lane = threadIdx.x & 31;
    const int wave = threadIdx.x >> 5;
    const int plane = blockIdx.y;              // 0..287
    const int tensor = plane / PLANES_PER_TENSOR;
    const int within = plane % PLANES_PER_TENSOR;

    const float* X =
        (tensor == 0 ? Xa : (tensor == 1 ? Xp : Xn)) + (size_t)within * PLANE_ELEMS;
    float* zr = Zr + (size_t)plane * PLANE_ELEMS;
    float* zi = Zi + (size_t)plane * PLANE_ELEMS;

    const int tile = blockIdx.x * 8 + wave;    // 0..255
    const int tm = tile >> 4;
    const int tn = tile & 15;
    const int hi = lane >> 4;                  // half-wave select (K +0 / +2)
    const int row = tm * 16 + (lane & 15);     // output row (M)
    const int col = tn * 16 + (lane & 15);     // output col (N)

    v8f cr = {}, ci = {};
    for (int k = 0; k < N256; k += 4) {
        const int ka = k + 2 * hi;
        // A fragments: rows of Fr/Fi (row-major, contiguous K pair)
        v2f ar = *(const v2f*)(Fr + row * N256 + ka);
        v2f ai = *(const v2f*)(Fi + row * N256 + ka);
        // B fragment: rows ka, ka+1 of X at column `col`
        v2f b;
        b.x = X[ka * N256 + col];
        b.y = X[(ka + 1) * N256 + col];
        cr = wmma_f32(ar, b, cr);
        ci = wmma_f32(ai, b, ci);
    }
#pragma unroll
    for (int v = 0; v < 8; ++v) {
        const int m = tm * 16 + v + 8 * hi;
        zr[m * N256 + col] = cr[v];
        zi[m * N256 + col] = ci[v];
    }
}

// ---------------------------------------------------------------------------
// Kernel 2: column transform Y = Z * F for a[i], p[i], n[i], n[j], fused with
// the |.| reductions. grid = (96, 32), block = 256 (8 waves); one wave per
// (c, 16x16 tile) of batch i. Writes deterministic per-tile partial sums.
// part layout: part[d][i][w], d in {ap, an_i, an_j}, w = c*256 + tile.
// ---------------------------------------------------------------------------
__global__ void fcr_stage2(const float* __restrict__ Zr,
                           const float* __restrict__ Zi,
                           const float* __restrict__ Fr,
                           const float* __restrict__ Fi,
                           const int* __restrict__ neg_idx,
                           float* __restrict__ part) {
    const int lane = threadIdx.x & 31;
    const int wave = threadIdx.x >> 5;
    const int i = blockIdx.y;                  // batch
    const int w = blockIdx.x * 8 + wave;       // 0..767
    const int c = w >> 8;                      // channel 0..2
    const int tile = w & 255;
    const int tm = tile >> 4;
    const int tn = tile & 15;
    const int hi = lane >> 4;
    const int row = tm * 16 + (lane & 15);
    const int col = tn * 16 + (lane & 15);

    const int j = neg_idx[i * 2 + 1];          // gathered negative batch

    const size_t P = PLANE_ELEMS;
    const float* Zar = Zr + ((size_t)(0 * PLANES_PER_TENSOR + i * 3 + c)) * P;
    const float* Zai = Zi + ((size_t)(0 * PLANES_PER_TENSOR + i * 3 + c)) * P;
    const float* Zpr = Zr + ((size_t)(1 * PLANES_PER_TENSOR + i * 3 + c)) * P;
    const float* Zpi = Zi + ((size_t)(1 * PLANES_PER_TENSOR + i * 3 + c)) * P;
    const float* Znr = Zr + ((size_t)(2 * PLANES_PER_TENSOR + i * 3 + c)) * P;
    const float* Zni = Zi + ((size_t)(2 * PLANES_PER_TENSOR + i * 3 + c)) * P;
    const float* Zmr = Zr + ((size_t)(2 * PLANES_PER_TENSOR + j * 3 + c)) * P;
    const float* Zmi = Zi + ((size_t)(2 * PLANES_PER_TENSOR + j * 3 + c)) * P;

    v8f yar = {}, yai = {}, ypr = {}, ypi = {};
    v8f ynr = {}, yni = {}, ymr = {}, ymi = {};

    const int rowOff = row * N256;
    const int colOff = col * N256;             // F symmetric: B[k][col]=F[col][k]

    for (int k = 0; k < N256; k += 4) {
        const int ka = k + 2 * hi;
        // B fragments (shared by all four planes): contiguous via F = F^T
        v2f br = *(const v2f*)(Fr + colOff + ka);
        v2f bi = *(const v2f*)(Fi + colOff + ka);
        v2f nbi;                               // -Fi fragment (f32 WMMA has no
        nbi.x = -bi.x;                         //  A/B NEG modifier, so negate
        nbi.y = -bi.y;                         //  once, shared by all planes)
        // A fragments: rows of Z (row-major, contiguous K pair)
        v2f azr = *(const v2f*)(Zar + rowOff + ka);
        v2f azi = *(const v2f*)(Zai + rowOff + ka);
        v2f pzr = *(const v2f*)(Zpr + rowOff + ka);
        v2f pzi = *(const v2f*)(Zpi + rowOff + ka);
        v2f nzr = *(const v2f*)(Znr + rowOff + ka);
        v2f nzi = *(const v2f*)(Zni + rowOff + ka);
        v2f mzr = *(const v2f*)(Zmr + rowOff + ka);
        v2f mzi = *(const v2f*)(Zmi + rowOff + ka);

        // Yr = Zr*Fr + Zi*(-Fi) ; Yi = Zr*Fi + Zi*Fr
        yar = wmma_f32(azr, br, yar);
        yar = wmma_f32(azi, nbi, yar);
        yai = wmma_f32(azr, bi, yai);
        yai = wmma_f32(azi, br, yai);

        ypr = wmma_f32(pzr, br, ypr);
        ypr = wmma_f32(pzi, nbi, ypr);
        ypi = wmma_f32(pzr, bi, ypi);
        ypi = wmma_f32(pzi, br, ypi);

        ynr = wmma_f32(nzr, br, ynr);
        ynr = wmma_f32(nzi, nbi, ynr);
        yni = wmma_f32(nzr, bi, yni);
        yni = wmma_f32(nzi, br, yni);

        ymr = wmma_f32(mzr, br, ymr);
        ymr = wmma_f32(mzi, nbi, ymr);
        ymi = wmma_f32(mzr, bi, ymi);
        ymi = wmma_f32(mzi, br, ymi);
    }

    float s_ap = 0.f, s_an = 0.f, s_am = 0.f;
#pragma unroll
    for (int v = 0; v < 8; ++v) {
        float dr = yar[v] - ypr[v], di = yai[v] - ypi[v];
        s_ap += sqrtf(dr * dr + di * di);
        dr = yar[v] - ynr[v]; di = yai[v] - yni[v];
        s_an += sqrtf(dr * dr + di * di);
        dr = yar[v] - ymr[v]; di = yai[v] - ymi[v];
        s_am += sqrtf(dr * dr + di * di);
    }
#pragma unroll
    for (int off = 16; off >= 1; off >>= 1) {
        s_ap += __shfl_down(s_ap, off, 32);
        s_an += __shfl_down(s_an, off, 32);
        s_am += __shfl_down(s_am, off, 32);
    }
    if (lane == 0) {
        const size_t base = (size_t)i * PART_PER_BATCH + w;
        part[0 * 32 * PART_PER_BATCH + base] = s_ap;
        part[1 * 32 * PART_PER_BATCH + base] = s_an;
        part[2 * 32 * PART_PER_BATCH + base] = s_am;
    }
}

// ---------------------------------------------------------------------------
// Kernel 3: final reduction to the contrastive scalar. Single block.
// ---------------------------------------------------------------------------
__global__ void fcr_finalize(const float* __restrict__ part,
                             float* __restrict__ out) {
    __shared__ float sums[96];                 // [d][i] flattened (d*32+i)
    const int t = threadIdx.x;
    if (t < 96) {
        const float* p = part + (size_t)t * PART_PER_BATCH;
        float s = 0.f;
        for (int k = 0; k < PART_PER_BATCH; ++k) s += p[k];
        sums[t] = s;
    }
    __syncthreads();
    if (t == 0) {
        const float inv = 1.0f / (3.0f * 256.0f * 256.0f);  // mean over (C,H,W)
        float acc = 0.f;
        for (int i = 0; i < 32; ++i) {
            float dap = sums[0 * 32 + i] * inv;
            float dan = sums[1 * 32 + i] * inv;
            float dam = sums[2 * 32 + i] * inv;
            acc += dap / (dan + EPS_F) + dap / (dam + EPS_F);
        }
        out[0] = acc / 64.0f;                  // / (MULTI_N_NUM * B)
    }
}

// ---------------------------------------------------------------------------
extern "C" void kernel_launch(void* const* d_in, const int* in_sizes, int n_in,
                              void* d_out, int out_size, void* d_ws, size_t ws_size,
                              hipStream_t stream) {
    const float* a = (const float*)d_in[0];
    const float* p = (const float*)d_in[1];
    const float* n = (const float*)d_in[2];
    const int* neg_idx = (const int*)d_in[3];

    float* ws = (float*)d_ws;
    float* Fr = ws;                                    // 65536
    float* Fi = Fr + PLANE_ELEMS;                      // 65536
    float* Zr = Fi + PLANE_ELEMS;                      // 288 * 65536
    float* Zi = Zr + (size_t)TOTAL_PLANES * PLANE_ELEMS;
    float* part = Zi + (size_t)TOTAL_PLANES * PLANE_ELEMS;  // 3*32*768

    fcr_twiddles<<<dim3(256), dim3(256), 0, stream>>>(Fr, Fi);
    fcr_stage1<<<dim3(32, TOTAL_PLANES), dim3(256), 0, stream>>>(a, p, n, Fr, Fi,
                                                                 Zr, Zi);
    fcr_stage2<<<dim3(96, 32), dim3(256), 0, stream>>>(Zr, Zi, Fr, Fi, neg_idx,
                                                       part);
    fcr_finalize<<<1, 256, 0, stream>>>(part, (float*)d_out);
}